// MultiResHashEncoder_85607288144208
// MI455X (gfx1250) — compile-verified
//
#include <hip/hip_runtime.h>
#include <math.h>

namespace {

constexpr int      kLevels   = 16;
constexpr int      kHashSize = 1 << 19;
constexpr unsigned kMask     = (unsigned)(kHashSize - 1);
constexpr unsigned kP0       = 1540863946u;
constexpr unsigned kP1       = 1257487969u;
constexpr unsigned kP2       = 1034312349u;

typedef __attribute__((ext_vector_type(2))) float f32x2;

struct ResPack { float r[kLevels]; };

__global__ __launch_bounds__(256) void hashgrid_encode(
    const float* __restrict__ x,
    const float* __restrict__ tables,
    float* __restrict__ out,
    ResPack rp,
    int total)
{
  const int tid = blockIdx.x * 256 + (int)threadIdx.x;
  if (tid >= total) return;

  const int lvl = tid & (kLevels - 1);   // lanes 0..15 -> levels of point n, 16..31 -> point n+1
  const int n   = tid >> 4;

  // 16 lanes share each address -> coalesced to one cacheline request per wave
  const float px = x[3 * n + 0];
  const float py = x[3 * n + 1];
  const float pz = x[3 * n + 2];
  const float res = rp.r[lvl];

  // Match reference rounding exactly: XLA computes x*res and -0.5 as two
  // separately-rounded ops; forbid FMA contraction here (compiler otherwise
  // emits v_pk_fma_f32 with -0.5 fused).
  float sx, sy, sz;
  {
#pragma clang fp contract(off)
    sx = px * res - 0.5f;
    sy = py * res - 0.5f;
    sz = pz * res - 0.5f;
  }

  const float bxf = floorf(sx), byf = floorf(sy), bzf = floorf(sz);
  const float fx = sx - bxf, fy = sy - byf, fz = sz - bzf;

  // Low 19 bits of int64 (coord * prime) == low bits of wrapping u32 multiply,
  // and remainder(h, 2^19) == h & 0x7FFFF (power-of-two floor-mod).
  const unsigned hx0 = (unsigned)(int)bxf * kP0, hx1 = hx0 + kP0;
  const unsigned hy0 = (unsigned)(int)byf * kP1, hy1 = hy0 + kP1;
  const unsigned hz0 = (unsigned)(int)bzf * kP2, hz1 = hz0 + kP2;

  const f32x2* __restrict__ tbl =
      reinterpret_cast<const f32x2*>(tables) + (size_t)lvl * (size_t)kHashSize;

  // idx bits: (ox, oy, oz) matching reference corner enumeration c = (ox<<2)|(oy<<1)|oz
  const unsigned i000 = (hx0 ^ hy0 ^ hz0) & kMask;
  const unsigned i001 = (hx0 ^ hy0 ^ hz1) & kMask;
  const unsigned i010 = (hx0 ^ hy1 ^ hz0) & kMask;
  const unsigned i011 = (hx0 ^ hy1 ^ hz1) & kMask;
  const unsigned i100 = (hx1 ^ hy0 ^ hz0) & kMask;
  const unsigned i101 = (hx1 ^ hy0 ^ hz1) & kMask;
  const unsigned i110 = (hx1 ^ hy1 ^ hz0) & kMask;
  const unsigned i111 = (hx1 ^ hy1 ^ hz1) & kMask;

  // 8 independent L2-resident b64 gathers; issued back-to-back for MLP.
  const f32x2 g000 = tbl[i000];
  const f32x2 g001 = tbl[i001];
  const f32x2 g010 = tbl[i010];
  const f32x2 g011 = tbl[i011];
  const f32x2 g100 = tbl[i100];
  const f32x2 g101 = tbl[i101];
  const f32x2 g110 = tbl[i110];
  const f32x2 g111 = tbl[i111];

  const float wx1 = fx, wx0 = 1.0f - fx;
  const float wy1 = fy, wy0 = 1.0f - fy;
  const float wz1 = fz, wz0 = 1.0f - fz;

  // Accumulate in reference corner order c = 0..7
  float a0, a1;
  { const float w = (wx0 * wy0) * wz0; a0 = w * g000.x;           a1 = w * g000.y;           }
  { const float w = (wx0 * wy0) * wz1; a0 = fmaf(w, g001.x, a0);  a1 = fmaf(w, g001.y, a1);  }
  { const float w = (wx0 * wy1) * wz0; a0 = fmaf(w, g010.x, a0);  a1 = fmaf(w, g010.y, a1);  }
  { const float w = (wx0 * wy1) * wz1; a0 = fmaf(w, g011.x, a0);  a1 = fmaf(w, g011.y, a1);  }
  { const float w = (wx1 * wy0) * wz0; a0 = fmaf(w, g100.x, a0);  a1 = fmaf(w, g100.y, a1);  }
  { const float w = (wx1 * wy0) * wz1; a0 = fmaf(w, g101.x, a0);  a1 = fmaf(w, g101.y, a1);  }
  { const float w = (wx1 * wy1) * wz0; a0 = fmaf(w, g110.x, a0);  a1 = fmaf(w, g110.y, a1);  }
  { const float w = (wx1 * wy1) * wz1; a0 = fmaf(w, g111.x, a0);  a1 = fmaf(w, g111.y, a1);  }

  // Streaming 256MB output: non-temporal b64 store (coalesced 256B per wave),
  // keeps the 64MB table working set resident in the 192MB L2.
  f32x2 o; o.x = a0; o.y = a1;
  __builtin_nontemporal_store(o, reinterpret_cast<f32x2*>(out) + tid);
}

} // namespace

extern "C" void kernel_launch(void* const* d_in, const int* in_sizes, int n_in,
                              void* d_out, int out_size, void* d_ws, size_t ws_size,
                              hipStream_t stream) {
  const float* x      = (const float*)d_in[0];
  const float* tables = (const float*)d_in[1];
  float*       out    = (float*)d_out;

  const int N = in_sizes[0] / 3;

  // Reproduce the reference's resolution table with identical double-precision
  // libm sequence (exp/log/pow), so floor() boundaries (levels 3/6/9/12/15)
  // match exactly. Deterministic; passed by value (graph-capture safe).
  ResPack rp;
  const double b = exp((log(512.0) - log(16.0)) / 15.0);
  for (int l = 0; l < kLevels; ++l) {
    int r = (int)floor(16.0 * pow(b, (double)l));
    if (r < 1) r = 1;
    rp.r[l] = (float)r;
  }

  const int total = N * kLevels;           // 33,554,432 threads: one per (point, level)
  const int block = 256;                   // 8 wave32s per block
  const int grid  = (total + block - 1) / block;

  hipLaunchKernelGGL(hashgrid_encode, dim3(grid), dim3(block), 0, stream,
                     x, tables, out, rp, total);
}